// SimpleHeteroGATPyG_74431783240150
// MI455X (gfx1250) — compile-verified
//
#include <hip/hip_runtime.h>
#include <math.h>

// ---------------------------------------------------------------------------
// SimpleHeteroGATPyG for MI455X (gfx1250, wave32).
//
// Dense GEMMs use V_WMMA_F32_16X16X4_F32 (exact fp32 semantics); each wave
// owns a 16x64 C strip so the A fragment is reused across 4 WMMAs per K-step.
// Edge phase: 3-pass segment softmax (atomic max via order-preserving uint
// encoding, atomic sum, then alpha-weighted scatter-add of 256-ch messages,
// gathered as float4 / global_load_b128).
//
// Attention scalars a = sum_c (h@lin)[:,h*64+c]*att[h,c] are computed as
// a = h @ W_fold with W_fold[k,h] = sum_c lin[k,h*64+c]*att[h,c], removing two
// 50000x64x256 GEMMs whose outputs were only reduced.
// ---------------------------------------------------------------------------

typedef float v2f __attribute__((ext_vector_type(2)));
typedef float v8f __attribute__((ext_vector_type(8)));

#define HID 64
#define HEADS 4
#define CH 256   // HEADS*HID

// ---------------------------------------------------------------------------
// fp32 WMMA GEMM: C[M,N] = A[M,K] @ B[K,N] (+ bias[N] if non-null).
// Row-major. Requires M%16==0, N%64==0, K%4==0.
// One wave computes a 16x64 strip of C (4 x 16x16 WMMA tiles, A frag reused).
// grid = (M/16, N/64), block = 32.
// ---------------------------------------------------------------------------
__global__ __launch_bounds__(32)
void wmma_gemm_f32(const float* __restrict__ A, const float* __restrict__ B,
                   const float* __restrict__ bias, float* __restrict__ C,
                   int M, int N, int K) {
  const int m0   = blockIdx.x << 4;
  const int n0   = blockIdx.y << 6;
  const int lane = threadIdx.x & 31;
  const int half = lane >> 4;   // 0: lanes 0-15, 1: lanes 16-31
  const int r    = lane & 15;

  v8f acc0 = {}, acc1 = {}, acc2 = {}, acc3 = {};
  const float* Arow = A + (long)(m0 + r) * K;   // A frag: M = r per lane
  for (int k0 = 0; k0 < K; k0 += 4) {
    // A 16x4 frag (ISA 7.12.2): VGPR0 = K{0|2}, VGPR1 = K{1|3} by lane half
    v2f a;
    a.x = Arow[k0 + 2 * half + 0];
    a.y = Arow[k0 + 2 * half + 1];
    // B 4x16 frags: row = K, col = lane&15; same K split across halves
    const float* B0 = B + (long)(k0 + 2 * half + 0) * N + n0 + r;
    const float* B1 = B + (long)(k0 + 2 * half + 1) * N + n0 + r;
    v2f b0, b1, b2, b3;
    b0.x = B0[0];  b0.y = B1[0];
    b1.x = B0[16]; b1.y = B1[16];
    b2.x = B0[32]; b2.y = B1[32];
    b3.x = B0[48]; b3.y = B1[48];
    acc0 = __builtin_amdgcn_wmma_f32_16x16x4_f32(false, a, false, b0, (short)0, acc0, false, false);
    acc1 = __builtin_amdgcn_wmma_f32_16x16x4_f32(false, a, false, b1, (short)0, acc1, false, false);
    acc2 = __builtin_amdgcn_wmma_f32_16x16x4_f32(false, a, false, b2, (short)0, acc2, false, false);
    acc3 = __builtin_amdgcn_wmma_f32_16x16x4_f32(false, a, false, b3, (short)0, acc3, false, false);
  }
  // C/D frag: VGPR j -> row m0 + j + 8*half, col n0 + nt*16 + r
  const v8f* accs[4] = {&acc0, &acc1, &acc2, &acc3};
#pragma unroll
  for (int nt = 0; nt < 4; ++nt) {
    const int col = n0 + nt * 16 + r;
    const float bv = bias ? bias[col] : 0.0f;
    const v8f av = *accs[nt];
#pragma unroll
    for (int j = 0; j < 8; ++j)
      C[(long)(m0 + j + 8 * half) * N + col] = av[j] + bv;
  }
}

// ---------------------------------------------------------------------------
// Fold att[HEADS,HID] into lin[HID, HEADS*HID]: w[k*4+h] = sum_c lin[k][h*64+c]*att[h][c]
// 1 block x 256 threads (k = t>>2, h = t&3).
// ---------------------------------------------------------------------------
__global__ void fold_att(const float* __restrict__ lin, const float* __restrict__ att,
                         float* __restrict__ w) {
  const int t = threadIdx.x;
  const int k = t >> 2, h = t & 3;
  float s = 0.f;
#pragma unroll 8
  for (int c = 0; c < HID; ++c) s += lin[k * CH + h * HID + c] * att[h * HID + c];
  w[k * 4 + h] = s;
}

// a[n*4+h] = sum_k H[n,k] * W[k*4+h]
__global__ void att_score(const float* __restrict__ H, const float* __restrict__ W,
                          float* __restrict__ out, int Nn) {
  const int t = blockIdx.x * blockDim.x + threadIdx.x;
  if (t >= Nn * 4) return;
  const int n = t >> 2, h = t & 3;
  const float* hr = H + (long)n * HID;
  float s = 0.f;
#pragma unroll 8
  for (int k = 0; k < HID; ++k) s += hr[k] * W[k * 4 + h];
  out[t] = s;
}

// Order-preserving float <-> uint encoding for atomic max over signed floats.
__device__ __forceinline__ unsigned enc_f(float f) {
  unsigned u = __float_as_uint(f);
  return (u & 0x80000000u) ? ~u : (u | 0x80000000u);
}
__device__ __forceinline__ float dec_f(unsigned u) {
  return __uint_as_float((u & 0x80000000u) ? (u & 0x7FFFFFFFu) : ~u);
}

__device__ __forceinline__ float lrelu(float x) { return x >= 0.f ? x : 0.2f * x; }

// Pass 1: per-destination, per-head max of e = lrelu(a_s[src]+a_d[dst]).
__global__ void edge_max_kernel(const int* __restrict__ src, const int* __restrict__ dst,
                                const float* __restrict__ a_s, const float* __restrict__ a_d,
                                unsigned* __restrict__ m, int E4) {
  const int t = blockIdx.x * blockDim.x + threadIdx.x;
  if (t >= E4) return;
  const int e = t >> 2, h = t & 3;
  const int s = src[e], d = dst[e];
  const float x = lrelu(a_s[s * 4 + h] + a_d[d * 4 + h]);
  atomicMax(&m[d * 4 + h], enc_f(x));
}

// Pass 2: softmax denominator.
__global__ void edge_sum_kernel(const int* __restrict__ src, const int* __restrict__ dst,
                                const float* __restrict__ a_s, const float* __restrict__ a_d,
                                const unsigned* __restrict__ m, float* __restrict__ ssum,
                                int E4) {
  const int t = blockIdx.x * blockDim.x + threadIdx.x;
  if (t >= E4) return;
  const int e = t >> 2, h = t & 3;
  const int s = src[e], d = dst[e];
  const float x = lrelu(a_s[s * 4 + h] + a_d[d * 4 + h]);
  atomicAdd(&ssum[d * 4 + h], expf(x - dec_f(m[d * 4 + h])));
}

// Pass 3: acc[dst] += alpha * HS[src].
// 64 threads per edge; thread t owns channels 4t..4t+3 (all in head t>>4);
// HS row gathered as float4 (global_load_b128), scatter via f32 atomics.
__global__ __launch_bounds__(64)
void edge_scatter_kernel(const int* __restrict__ src, const int* __restrict__ dst,
                         const float* __restrict__ a_s, const float* __restrict__ a_d,
                         const unsigned* __restrict__ m, const float* __restrict__ ssum,
                         const float* __restrict__ HS, float* __restrict__ acc) {
  const int e = blockIdx.x;
  const int t = threadIdx.x;     // 0..63
  const int h = t >> 4;          // head (uniform per 16 threads)
  const int s = src[e], d = dst[e];
  const float x = lrelu(a_s[s * 4 + h] + a_d[d * 4 + h]);
  const float alpha = expf(x - dec_f(m[d * 4 + h])) / (ssum[d * 4 + h] + 1e-16f);
  const float4 v = *(const float4*)(HS + (long)s * CH + t * 4);
  float* ap = acc + (long)d * CH + t * 4;
  atomicAdd(ap + 0, v.x * alpha);
  atomicAdd(ap + 1, v.y * alpha);
  atomicAdd(ap + 2, v.z * alpha);
  atomicAdd(ap + 3, v.w * alpha);
}

// buf[t] = elu(buf[t] + b1[c] + b2[c])   (b2 may be null)
__global__ void bias_elu_kernel(float* __restrict__ buf, const float* __restrict__ b1,
                                const float* __restrict__ b2, long n) {
  const long t = (long)blockIdx.x * blockDim.x + threadIdx.x;
  if (t >= n) return;
  const int c = (int)(t & (CH - 1));
  float x = buf[t] + b1[c] + (b2 ? b2[c] : 0.f);
  buf[t] = x > 0.f ? x : expm1f(x);
}

// ---------------------------------------------------------------------------
extern "C" void kernel_launch(void* const* d_in, const int* in_sizes, int n_in,
                              void* d_out, int out_size, void* d_ws, size_t ws_size,
                              hipStream_t stream) {
  const float* x_user = (const float*)d_in[0];
  const float* x_post = (const float*)d_in[1];
  const int* src_e[3] = {(const int*)d_in[2], (const int*)d_in[4], (const int*)d_in[6]};
  const int* dst_e[3] = {(const int*)d_in[3], (const int*)d_in[5], (const int*)d_in[7]};
  const float* W_proj = (const float*)d_in[8];
  const float* b_proj = (const float*)d_in[9];
  const float* lin[3]   = {(const float*)d_in[10], (const float*)d_in[14], (const float*)d_in[18]};
  const float* att_s[3] = {(const float*)d_in[11], (const float*)d_in[15], (const float*)d_in[19]};
  const float* att_d[3] = {(const float*)d_in[12], (const float*)d_in[16], (const float*)d_in[20]};
  const float* bias_e[3]= {(const float*)d_in[13], (const float*)d_in[17], (const float*)d_in[21]};
  const float* W_out = (const float*)d_in[22];
  const float* b_out = (const float*)d_in[23];

  const int NU = in_sizes[0] / 128;   // 50000 (divisible by 16)
  const int NP = in_sizes[1] / 128;   // 50000
  const int E  = in_sizes[2];         // 400000
  const int NMAX = NU > NP ? NU : NP;

  // ---- workspace layout (floats), ~131 MB ----
  float* ws = (float*)d_ws;
  size_t off = 0;
  float* hu    = ws + off; off += (size_t)NU * HID;
  float* hp    = ws + off; off += (size_t)NP * HID;
  float* HS    = ws + off; off += (size_t)NMAX * CH;   // reused per edge type
  float* acc_u = ws + off; off += (size_t)NU * CH;
  float* a_s   = ws + off; off += (size_t)NMAX * 4;
  float* a_d   = ws + off; off += (size_t)NMAX * 4;
  unsigned* mx = (unsigned*)(ws + off); off += (size_t)NMAX * 4;
  float* ssum  = ws + off; off += (size_t)NMAX * 4;
  float* w_as  = ws + off; off += 256;
  float* w_ad  = ws + off; off += 256;

  float* out_user = (float*)d_out;                   // [NU, 64]
  float* out_post = (float*)d_out + (size_t)NU * 64; // [NP, 256] (accumulated here)

  // ---- 1) input projection (WMMA, fp32): h = x @ W_proj + b_proj ----
  wmma_gemm_f32<<<dim3(NU / 16, HID / 64), 32, 0, stream>>>(x_user, W_proj, b_proj, hu, NU, HID, 128);
  wmma_gemm_f32<<<dim3(NP / 16, HID / 64), 32, 0, stream>>>(x_post, W_proj, b_proj, hp, NP, HID, 128);

  // ---- 2) zero accumulators ----
  hipMemsetAsync(acc_u, 0, (size_t)NU * CH * sizeof(float), stream);
  hipMemsetAsync(out_post, 0, (size_t)NP * CH * sizeof(float), stream);

  // ---- 3) edge types, sequential (HS / a / m / s buffers reused) ----
  // et 0: u2u (hu -> hu, acc_u); et 1: p2u (hp -> hu, acc_u); et 2: u2p (hu -> hp, out_post)
  const float* hsrc_t[3] = {hu, hp, hu};
  const float* hdst_t[3] = {hu, hu, hp};
  const int ns_t[3] = {NU, NP, NU};
  const int nd_t[3] = {NU, NU, NP};
  float* acc_t[3] = {acc_u, acc_u, out_post};

  for (int t = 0; t < 3; ++t) {
    const int ns = ns_t[t], nd = nd_t[t];
    // fold attention vectors into [64,4] matrices
    fold_att<<<1, 256, 0, stream>>>(lin[t], att_s[t], w_as);
    fold_att<<<1, 256, 0, stream>>>(lin[t], att_d[t], w_ad);
    // source features through lin (WMMA): HS = hsrc @ lin  [ns, 256]
    wmma_gemm_f32<<<dim3(ns / 16, CH / 64), 32, 0, stream>>>(
        hsrc_t[t], lin[t], nullptr, HS, ns, CH, HID);
    // attention scalars
    att_score<<<(ns * 4 + 255) / 256, 256, 0, stream>>>(hsrc_t[t], w_as, a_s, ns);
    att_score<<<(nd * 4 + 255) / 256, 256, 0, stream>>>(hdst_t[t], w_ad, a_d, nd);
    // segment softmax state
    hipMemsetAsync(mx, 0, (size_t)nd * 4 * sizeof(unsigned), stream);  // 0 < enc(any real) -> identity for max
    hipMemsetAsync(ssum, 0, (size_t)nd * 4 * sizeof(float), stream);
    const int E4 = E * 4;
    edge_max_kernel<<<(E4 + 255) / 256, 256, 0, stream>>>(src_e[t], dst_e[t], a_s, a_d, mx, E4);
    edge_sum_kernel<<<(E4 + 255) / 256, 256, 0, stream>>>(src_e[t], dst_e[t], a_s, a_d, mx, ssum, E4);
    edge_scatter_kernel<<<E, 64, 0, stream>>>(src_e[t], dst_e[t], a_s, a_d, mx, ssum, HS, acc_t[t]);
  }

  // ---- 4) bias + ELU ----
  {
    long n_u = (long)NU * CH, n_p = (long)NP * CH;
    bias_elu_kernel<<<(int)((n_u + 255) / 256), 256, 0, stream>>>(acc_u, bias_e[0], bias_e[1], n_u);
    bias_elu_kernel<<<(int)((n_p + 255) / 256), 256, 0, stream>>>(out_post, bias_e[2], nullptr, n_p);
  }

  // ---- 5) output projection for users (WMMA): out_user = elu(acc_u) @ W_out + b_out ----
  wmma_gemm_f32<<<dim3(NU / 16, 64 / 64), 32, 0, stream>>>(acc_u, W_out, b_out, out_user, NU, 64, CH);
}